// MaskHeadProposals_70901320122419
// MI455X (gfx1250) — compile-verified
//
#include <hip/hip_runtime.h>
#include <hip/hip_bf16.h>
#include <stdint.h>

// Problem constants (from reference)
#define B_IMG   8
#define G_GT    64
#define P_PROP  2000
#define N_BOX   (G_GT + P_PROP)   // 2064
#define NPAD    4096              // next pow2 for bitonic sort
#define MAXI    320
#define NTHR    512               // 16 wave32 waves
#define NMS_THR_F 0.3f

// ---- CDNA5 async global->LDS DMA (ASYNCcnt-tracked, per-lane addresses) ----
__device__ __forceinline__ void async_copy_b128(uint32_t lds_byte_addr, const void* gptr) {
    // GLOBAL_LOAD_ASYNC_TO_LDS_B128: vdst = LDS byte address VGPR, vaddr = 64-bit global addr
    asm volatile("global_load_async_to_lds_b128 %0, %1, off"
                 :: "v"(lds_byte_addr), "v"((unsigned long long)(uintptr_t)gptr)
                 : "memory");
}
__device__ __forceinline__ void wait_asynccnt0() {
    asm volatile("s_wait_asynccnt 0" ::: "memory");
}

__global__ __launch_bounds__(NTHR) void nms_select_kernel(
    const float* __restrict__ cls_prop,   // (B,P)
    const float* __restrict__ gt_cls,     // (B,G)   (also the GT scores)
    const float* __restrict__ box_prop,   // (B,P,4)
    const float* __restrict__ gt_box,     // (B,G,4)
    const float* __restrict__ scr_prop,   // (B,P)
    float* __restrict__ out)              // cls(B,320) | box(B,320,4) | scr(B,320)
{
    // sKeys (33,792B) is reused as the sorted-box array (33,024B) after the sort.
    __shared__ __align__(16) unsigned long long sKeys[4224];
    __shared__ uint32_t      sIdx[N_BOX];       // sorted position -> original index
    __shared__ unsigned char sKeepS[N_BOX];     // keep mask, sorted order
    __shared__ unsigned char sKeepO[N_BOX];     // keep mask, original order
    __shared__ uint32_t      sOutIdx[MAXI];
    __shared__ int           sCnt;

    float* sBoxes = reinterpret_cast<float*>(sKeys);   // [N_BOX][4], sorted order

    const int img = blockIdx.x;
    const int tid = threadIdx.x;

    const float* gtc = gt_cls   + (size_t)img * G_GT;
    const float* gtb = gt_box   + (size_t)img * G_GT * 4;
    const float* clp = cls_prop + (size_t)img * P_PROP;
    const float* bpp = box_prop + (size_t)img * P_PROP * 4;
    const float* spp = scr_prop + (size_t)img * P_PROP;

    // ---- Phase 1: build stable-descending sort keys ----------------------
    // key = (score_bits << 32) | (0xFFFFFFFF - index); scores >= 0 so IEEE
    // bits are monotonic; low word reproduces stable argsort tie-break.
    for (int n = tid; n < NPAD; n += NTHR) {
        unsigned long long key = 0ull;                 // pads sort last
        if (n < N_BOX) {
            float s = (n < G_GT) ? gtc[n] : spp[n - G_GT];
            uint32_t sb = __float_as_uint(s);
            key = ((unsigned long long)sb << 32) |
                  (uint32_t)(0xFFFFFFFFu - (uint32_t)n);
        }
        sKeys[n] = key;
    }

    // ---- Phase 2: bitonic sort (descending) ------------------------------
    for (int k = 2; k <= NPAD; k <<= 1) {
        for (int j = k >> 1; j > 0; j >>= 1) {
            __syncthreads();
            for (int i = tid; i < NPAD; i += NTHR) {
                int ixj = i ^ j;
                if (ixj > i) {
                    unsigned long long a = sKeys[i];
                    unsigned long long b = sKeys[ixj];
                    bool desc = ((i & k) == 0);
                    if (desc ? (a < b) : (a > b)) { sKeys[i] = b; sKeys[ixj] = a; }
                }
            }
        }
    }
    __syncthreads();

    // ---- Phase 3: extract sorted index order (before clobbering sKeys) ---
    for (int t = tid; t < N_BOX; t += NTHR) {
        sIdx[t]   = 0xFFFFFFFFu - (uint32_t)(sKeys[t] & 0xFFFFFFFFull);
        sKeepS[t] = 1;
    }
    __syncthreads();

    // ---- Phase 4: gather boxes in sorted order via async b128 DMA --------
    // Low 32 bits of a generic LDS pointer are the LDS byte offset (aperture
    // rules, ISA 10.2), which is what GLOBAL_LOAD_ASYNC_TO_LDS's VDST wants.
    {
        uint32_t lds_base = (uint32_t)(uintptr_t)sBoxes;
        for (int t = tid; t < N_BOX; t += NTHR) {
            uint32_t n = sIdx[t];
            const float* src = (n < G_GT) ? (gtb + (size_t)n * 4)
                                          : (bpp + (size_t)(n - G_GT) * 4);
            async_copy_b128(lds_base + (uint32_t)t * 16u, src);
        }
        wait_asynccnt0();       // each wave drains its own ASYNCcnt
    }
    __syncthreads();            // ...then all waves' DMAs are visible

    // ---- Phase 5: greedy NMS over sorted order ---------------------------
    for (int i = 0; i < N_BOX; ++i) {
        __syncthreads();
        if (!sKeepS[i]) continue;          // uniform read -> uniform branch
        float x1 = sBoxes[4*i+0], y1 = sBoxes[4*i+1];
        float x2 = sBoxes[4*i+2], y2 = sBoxes[4*i+3];
        float ai = (x2 - x1) * (y2 - y1);
        for (int jj = i + 1 + tid; jj < N_BOX; jj += NTHR) {
            if (!sKeepS[jj]) continue;
            float u1 = sBoxes[4*jj+0], v1 = sBoxes[4*jj+1];
            float u2 = sBoxes[4*jj+2], v2 = sBoxes[4*jj+3];
            float aj = (u2 - u1) * (v2 - v1);
            float xx1 = fmaxf(x1, u1), yy1 = fmaxf(y1, v1);
            float xx2 = fminf(x2, u2), yy2 = fminf(y2, v2);
            float inter = fmaxf(xx2 - xx1, 0.f) * fmaxf(yy2 - yy1, 0.f);
            float uni   = fmaxf(ai + aj - inter, 1e-9f);
            if (inter > NMS_THR_F * uni) sKeepS[jj] = 0;   // own byte only
        }
    }
    __syncthreads();

    // ---- Phase 6: scatter keep mask back to original index order ---------
    for (int t = tid; t < N_BOX; t += NTHR)
        sKeepO[sIdx[t]] = sKeepS[t];       // sIdx is a permutation
    __syncthreads();

    // ---- Phase 7: compact first MAXI kept indices (ascending) ------------
    if (tid == 0) {
        int c = 0;
        for (int n = 0; n < N_BOX; ++n)
            if (sKeepO[n] && c < MAXI) sOutIdx[c++] = (uint32_t)n;
        sCnt = c;
    }
    __syncthreads();

    // ---- Phase 8: write outputs (zero padded) ----------------------------
    float* out_cls = out;
    float* out_box = out + (size_t)B_IMG * MAXI;
    float* out_scr = out + (size_t)B_IMG * MAXI * 5;   // 320 + 320*4 per image

    int cnt = sCnt;
    for (int t = tid; t < MAXI; t += NTHR) {
        float cv = 0.f, sv = 0.f, b0 = 0.f, b1 = 0.f, b2 = 0.f, b3 = 0.f;
        if (t < cnt) {
            uint32_t n = sOutIdx[t];
            const float* bp;
            if (n < G_GT) { cv = gtc[n]; sv = cv;              bp = gtb + (size_t)n * 4; }
            else { uint32_t m = n - G_GT; cv = clp[m]; sv = spp[m]; bp = bpp + (size_t)m * 4; }
            b0 = bp[0]; b1 = bp[1]; b2 = bp[2]; b3 = bp[3];
        }
        out_cls[(size_t)img * MAXI + t] = cv;
        float* ob = out_box + ((size_t)img * MAXI + t) * 4;
        ob[0] = b0; ob[1] = b1; ob[2] = b2; ob[3] = b3;
        out_scr[(size_t)img * MAXI + t] = sv;
    }
}

extern "C" void kernel_launch(void* const* d_in, const int* in_sizes, int n_in,
                              void* d_out, int out_size, void* d_ws, size_t ws_size,
                              hipStream_t stream) {
    (void)in_sizes; (void)n_in; (void)d_ws; (void)ws_size; (void)out_size;
    const float* cls_prop = (const float*)d_in[0];
    const float* gt_cls   = (const float*)d_in[1];
    const float* box_prop = (const float*)d_in[2];
    const float* gt_box   = (const float*)d_in[3];
    const float* scr_prop = (const float*)d_in[4];

    nms_select_kernel<<<B_IMG, NTHR, 0, stream>>>(
        cls_prop, gt_cls, box_prop, gt_box, scr_prop, (float*)d_out);
}